// GraphSage_13039520710737
// MI455X (gfx1250) — compile-verified
//
#include <hip/hip_runtime.h>

typedef __attribute__((ext_vector_type(2))) float v2f;
typedef __attribute__((ext_vector_type(8))) float v8f;

#define D_FEAT   128
#define K_TOT    256
#define LDS_PITCH 260   // 260 % 64 == 4 -> bank-conflict-free column reads

// ---------------------------------------------------------------------------
// Zero workspace (h1 + degree). ws is poisoned by the harness, so every call
// must re-zero before the scatter phase.
// ---------------------------------------------------------------------------
__global__ void gs_zero_kernel(float* __restrict__ p, int n) {
    int i = blockIdx.x * blockDim.x + threadIdx.x;
    if (i < n) p[i] = 0.0f;
}

// ---------------------------------------------------------------------------
// Phase 1: one wave32 per edge.  Each lane carries 4 consecutive floats of the
// gathered x[dst] row (float4 = global_load_b128) and scatter-adds into
// h1[src] with fp32 global atomics.  Lane 0 accumulates the degree.
// ---------------------------------------------------------------------------
__global__ __launch_bounds__(256) void gs_scatter_kernel(
        const float* __restrict__ x,
        const int*   __restrict__ esrc,
        const int*   __restrict__ edst,
        const float* __restrict__ eval,
        float* __restrict__ h1,
        float* __restrict__ deg,
        int E) {
    int gtid = blockIdx.x * blockDim.x + threadIdx.x;
    int edge = gtid >> 5;
    int lane = threadIdx.x & 31;
    if (edge >= E) return;

    int   s = esrc[edge];
    int   d = edst[edge];
    float v = eval[edge];

    const float4* xr = (const float4*)(x + (size_t)d * D_FEAT);
    float4 m = xr[lane];

    float* hr = h1 + (size_t)s * D_FEAT + lane * 4;
    atomicAdd(hr + 0, m.x * v);
    atomicAdd(hr + 1, m.y * v);
    atomicAdd(hr + 2, m.z * v);
    atomicAdd(hr + 3, m.w * v);

    if (lane == 0) atomicAdd(deg + s, v);
}

// ---------------------------------------------------------------------------
// Phase 2: fused normalize + GEMM   out = [x , h1/(deg+1e-6)] @ W
// Block = 256 threads = 8 waves, one 16-row tile per block.
// Stage 16x256 fp32 activation tile into LDS (row pitch 260 for conflict-free
// column access), then each wave computes one 16x16 output tile with a chain
// of 64 V_WMMA_F32_16X16X4_F32 ops (exact fp32 matrix math).
// ---------------------------------------------------------------------------
__global__ __launch_bounds__(256) void gs_gemm_kernel(
        const float* __restrict__ x,
        const float* __restrict__ h1,
        const float* __restrict__ deg,
        const float* __restrict__ W,     // [256 x 128] row-major
        float* __restrict__ out,         // [N x 128]
        int N) {
    __shared__ float sh[16 * LDS_PITCH];

    const int r0  = blockIdx.x * 16;
    const int tid = threadIdx.x;

    // --- cooperative stage: 16 rows x 256 cols, normalization fused in -----
    for (int i = tid; i < 16 * K_TOT; i += 256) {
        int row = i >> 8;          // 0..15
        int col = i & 255;         // 0..255
        int gr  = r0 + row;
        float v = 0.0f;
        if (gr < N) {
            if (col < D_FEAT) {
                v = x[(size_t)gr * D_FEAT + col];
            } else {
                float dg = deg[gr];
                v = h1[(size_t)gr * D_FEAT + (col - D_FEAT)] * (1.0f / (dg + 1e-6f));
            }
        }
        sh[row * LDS_PITCH + col] = v;
    }
    __syncthreads();

    const int lane = tid & 31;
    const int wv   = tid >> 5;          // column tile 0..7 (16 cols each)
    const int m    = lane & 15;
    const int kb   = (lane >> 4) << 1;  // 0 or 2: K sub-offset per ISA layout

    v8f acc = {};
    const float* wp = W + (size_t)(wv * 16 + m);   // B col = wv*16 + (lane&15)

    #pragma unroll
    for (int k = 0; k < K_TOT; k += 4) {
        // A[16x4]: lane l holds A[m][kb], A[m][kb+1]
        const float* ap = &sh[m * LDS_PITCH + k + kb];
        v2f a = { ap[0], ap[1] };
        // B[4x16]: lane l holds B[kb][m], B[kb+1][m]  (rows of W)
        const float* bp = wp + (size_t)(k + kb) * D_FEAT;
        v2f b = { bp[0], bp[D_FEAT] };
        acc = __builtin_amdgcn_wmma_f32_16x16x4_f32(
                  /*neg_a=*/false, a, /*neg_b=*/false, b,
                  /*c_mod=*/(short)0, acc,
                  /*reuse_a=*/false, /*reuse_b=*/false);
    }

    // D layout: VGPR r -> row r + 8*(lane>>4), col lane&15
    const int rbase = r0 + 8 * (lane >> 4);
    const int ocol  = wv * 16 + m;
    #pragma unroll
    for (int r = 0; r < 8; ++r) {
        int orow = rbase + r;
        if (orow < N) out[(size_t)orow * D_FEAT + ocol] = acc[r];
    }
}

// ---------------------------------------------------------------------------
extern "C" void kernel_launch(void* const* d_in, const int* in_sizes, int n_in,
                              void* d_out, int out_size, void* d_ws, size_t ws_size,
                              hipStream_t stream) {
    const float* x    = (const float*)d_in[0];   // [N, 128]
    const float* W    = (const float*)d_in[1];   // [256, 128]
    const int*   esrc = (const int*)  d_in[2];   // [E]
    const int*   edst = (const int*)  d_in[3];   // [E]
    const float* eval = (const float*)d_in[4];   // [E]
    float*       out  = (float*)d_out;           // [N, 128]

    const int N = in_sizes[0] / D_FEAT;
    const int E = in_sizes[2];

    float* h1  = (float*)d_ws;                   // N * 128 floats
    float* deg = h1 + (size_t)N * D_FEAT;        // N floats

    // zero h1 + degree
    int nz = N * (D_FEAT + 1);
    gs_zero_kernel<<<(nz + 255) / 256, 256, 0, stream>>>(h1, nz);

    // scatter: one wave per edge, 8 edges per 256-thread block
    int nblk = (E + 7) / 8;
    gs_scatter_kernel<<<nblk, 256, 0, stream>>>(x, esrc, edst, eval, h1, deg, E);

    // fused normalize + WMMA GEMM: one 16-row tile per block
    int gblk = (N + 15) / 16;
    gs_gemm_kernel<<<gblk, 256, 0, stream>>>(x, h1, deg, W, out, N);
}